// DualStreamBiMambaBackbone_75960791598064
// MI455X (gfx1250) — compile-verified
//
#include <hip/hip_runtime.h>
#include <math.h>

// ---------------------------------------------------------------------------
// DualStreamBiMambaBackbone for MI455X (gfx1250, wave32, WMMA)
// All dense matmuls -> V_WMMA_F32_16X16X4_F32 (exact fp32 path).
// 4 independent mamba stacks fused into one super-batch per kernel.
// ---------------------------------------------------------------------------

constexpr int Bc  = 8;
constexpr int Lc  = 512;
constexpr int BLc = Bc * Lc;          // 4096 rows per stack
constexpr int Dm  = 64;               // stream model dim (D_LEN == D_STRUCT)
constexpr int Din = 128;              // EXPAND * Dm
constexpr int Ns  = 16;               // D_STATE
constexpr int DR  = 4;                // dt rank = ceil(64/16)
constexpr int XD  = DR + 2 * Ns;      // 36
constexpr int NSTK = 4;               // len_fwd, len_bwd, struct_fwd, struct_bwd

typedef __attribute__((ext_vector_type(2))) float v2f;
typedef __attribute__((ext_vector_type(8))) float v8f;

struct P4 { const float* v[4]; };     // per-stack parameter pointers (by value)

// ---------------------------------------------------------------------------
// Generic fp32 WMMA GEMM:  C[m,n] = sum_k A[m,k] * W[n,k] (+bias[n]) (+res[m,n])
// A: M x K row-major, W: N x K row-major (i.e. B = W^T), one wave per 16x16 tile.
// Stack selected by output row block (rowsPerStack rows per stack).
// ---------------------------------------------------------------------------
__global__ void k_gemm_wmma_ms(const float* __restrict__ A,
                               P4 W, P4 bias,
                               const float* __restrict__ res,
                               float* __restrict__ C,
                               int N, int K, int rowsPerStack, int hasBias)
{
    const int lane = threadIdx.x & 31;
    const int tn   = blockIdx.x;               // N tile
    const int tm   = blockIdx.y;               // M tile
    const int s    = (tm * 16) / rowsPerStack; // stack index
    const float* __restrict__ Wp = W.v[s];

    const int half = lane >> 4;                // 0: k pair {0,1}, 1: k pair {2,3}
    const int l16  = lane & 15;
    const int rowA = tm * 16 + l16;            // A-matrix row for this lane
    const int n    = tn * 16 + l16;            // output column for this lane

    v8f acc = {};
    for (int k = 0; k < K; k += 4) {
        const int kk = k + 2 * half;
        v2f av = *(const v2f*)(A  + (size_t)rowA * K + kk);  // A[m][kk], A[m][kk+1]
        v2f bv = *(const v2f*)(Wp + (size_t)n    * K + kk);  // B[kk][n], B[kk+1][n]
        acc = __builtin_amdgcn_wmma_f32_16x16x4_f32(
                  false, av, false, bv, (short)0, acc, false, false);
    }

    const float bn   = hasBias ? bias.v[s][n] : 0.0f;
    const int mBase  = tm * 16 + half * 8;     // lanes<16 hold rows 0..7, >=16 rows 8..15
#pragma unroll
    for (int j = 0; j < 8; ++j) {
        const int m = mBase + j;
        float vv = acc[j] + bn;
        if (res) vv += res[(size_t)m * N + n];
        C[(size_t)m * N + n] = vv;
    }
}

// ---------------------------------------------------------------------------
// Helpers
// ---------------------------------------------------------------------------
__device__ __forceinline__ float gelu_exact(float x) {
    return 0.5f * x * (1.0f + erff(x * 0.70710678118654752f));
}
__device__ __forceinline__ float sigmoidf_(float x) {
    return 1.0f / (1.0f + expf(-x));
}
__device__ __forceinline__ float pe_val(int l, int d) {
    const int i2 = (d >> 1) << 1;  // 2*(d/2)
    const float f = expf(-logf(10000.0f) * (float)i2 / (float)Dm);
    const float a = (float)l * f;
    return (d & 1) ? cosf(a) : sinf(a);
}

// ---------------------------------------------------------------------------
// Embedding: x -> x_len, x_struct  (block = 64 threads = one (b,l) row)
// ---------------------------------------------------------------------------
__global__ void k_embed(const float* __restrict__ X,
                        const float* lenW, const float* lenb,
                        const float* lng,  const float* lnbb,
                        const float* contW, const float* contb,
                        const float* cng,  const float* cnb,
                        const float* dir_emb,
                        float* __restrict__ xlen, float* __restrict__ xstr)
{
    __shared__ float red[64];
    const int bl = blockIdx.x;
    const int l  = bl % Lc;
    const int d  = threadIdx.x;
    const float x0 = X[bl * 4 + 0];
    const float x1 = X[bl * 4 + 1];
    const float x2 = X[bl * 4 + 2];
    const float pe = pe_val(l, d);

    // ---- length stream: gelu(ln(x0*W+b)) + pe
    float v = x0 * lenW[d] + lenb[d];
    red[d] = v; __syncthreads();
    for (int st = 32; st >= 1; st >>= 1) { if (d < st) red[d] += red[d + st]; __syncthreads(); }
    const float mean = red[0] * (1.0f / Dm); __syncthreads();
    const float c = v - mean;
    red[d] = c * c; __syncthreads();
    for (int st = 32; st >= 1; st >>= 1) { if (d < st) red[d] += red[d + st]; __syncthreads(); }
    const float var = red[0] * (1.0f / Dm); __syncthreads();
    xlen[(size_t)bl * Dm + d] =
        gelu_exact(c * rsqrtf(var + 1e-5f) * lng[d] + lnbb[d]) + pe;

    // ---- struct stream: gelu(ln(x2*W+b)) + dir_emb + pe
    float e = x2 * contW[d] + contb[d];
    red[d] = e; __syncthreads();
    for (int st = 32; st >= 1; st >>= 1) { if (d < st) red[d] += red[d + st]; __syncthreads(); }
    const float m2 = red[0] * (1.0f / Dm); __syncthreads();
    const float c2 = e - m2;
    red[d] = c2 * c2; __syncthreads();
    for (int st = 32; st >= 1; st >>= 1) { if (d < st) red[d] += red[d + st]; __syncthreads(); }
    const float v2 = red[0] * (1.0f / Dm);
    const int idx = (x1 > 0.0f) ? 1 : 0;
    xstr[(size_t)bl * Dm + d] =
        gelu_exact(c2 * rsqrtf(v2 + 1e-5f) * cng[d] + cnb[d]) + dir_emb[idx * Dm + d] + pe;
}

// Pack the 4 stack inputs: s0=xlen, s1=flip(xlen), s2=xstr, s3=flip(xstr)
__global__ void k_pack(const float* __restrict__ xlen, const float* __restrict__ xstr,
                       float* __restrict__ bufA)
{
    const int idx = blockIdx.x * blockDim.x + threadIdx.x;
    if (idx >= NSTK * BLc * Dm) return;
    const int s  = idx / (BLc * Dm);
    const int r  = idx - s * (BLc * Dm);
    const int bl = r / Dm;
    const int d  = r - bl * Dm;
    const int b  = bl / Lc;
    const int l  = bl % Lc;
    const float* src = (s < 2) ? xlen : xstr;
    const int sl = (s & 1) ? (Lc - 1 - l) : l;
    bufA[idx] = src[((size_t)b * Lc + sl) * Dm + d];
}

// LayerNorm over Dm=64, per-stack gamma/beta
__global__ void k_ln_ms(const float* __restrict__ in, P4 g, P4 b,
                        float* __restrict__ out, int rowsPerStack)
{
    __shared__ float red[64];
    const int row = blockIdx.x;                // 0 .. NSTK*BLc-1
    const int s   = row / rowsPerStack;
    const int d   = threadIdx.x;
    const float v = in[(size_t)row * Dm + d];
    red[d] = v; __syncthreads();
    for (int st = 32; st >= 1; st >>= 1) { if (d < st) red[d] += red[d + st]; __syncthreads(); }
    const float mean = red[0] * (1.0f / Dm); __syncthreads();
    const float c = v - mean;
    red[d] = c * c; __syncthreads();
    for (int st = 32; st >= 1; st >>= 1) { if (d < st) red[d] += red[d + st]; __syncthreads(); }
    const float var = red[0] * (1.0f / Dm);
    out[(size_t)row * Dm + d] = c * rsqrtf(var + 1e-5f) * g.v[s][d] + b.v[s][d];
}

// Depthwise causal conv (K=4) over the first Din columns of xz, then SiLU
__global__ void k_conv_silu_ms(const float* __restrict__ xz, P4 w, P4 cb,
                               float* __restrict__ xc)
{
    const int idx = blockIdx.x * blockDim.x + threadIdx.x;
    if (idx >= NSTK * BLc * Din) return;
    const int d  = idx % Din;
    const int bl = idx / Din;                 // global row over 4*BL
    const int s  = bl / BLc;
    const int t  = bl % Lc;
    const float* wp = w.v[s] + d * 4;
    float acc = cb.v[s][d];
#pragma unroll
    for (int j = 0; j < 4; ++j) {
        const int tt = t + j - 3;             // causal: pad (K-1) on the left
        if (tt >= 0)
            acc += wp[j] * xz[((size_t)(bl + tt - t)) * (2 * Din) + d];
    }
    xc[(size_t)bl * Din + d] = acc * sigmoidf_(acc);
}

// x_dbl = xc @ W_x^T  (N=36, K=128) — tiny, plain VALU
__global__ void k_xdbl_ms(const float* __restrict__ xc, P4 Wx, float* __restrict__ xdbl)
{
    const int idx = blockIdx.x * blockDim.x + threadIdx.x;
    if (idx >= NSTK * BLc * XD) return;
    const int c   = idx % XD;
    const int row = idx / XD;
    const int s   = row / BLc;
    const float* wr = Wx.v[s] + (size_t)c * Din;
    const float* xr = xc + (size_t)row * Din;
    float acc = 0.0f;
    for (int k = 0; k < Din; ++k) acc += xr[k] * wr[k];
    xdbl[(size_t)row * XD + c] = acc;
}

// dt = softplus(x_dbl[:, :DR] @ W_dt^T + b_dt)
__global__ void k_dt_ms(const float* __restrict__ xdbl, P4 Wdt, P4 bdt,
                        float* __restrict__ dt)
{
    const int idx = blockIdx.x * blockDim.x + threadIdx.x;
    if (idx >= NSTK * BLc * Din) return;
    const int d   = idx % Din;
    const int row = idx / Din;
    const int s   = row / BLc;
    const float* xr = xdbl + (size_t)row * XD;
    const float* wr = Wdt.v[s] + d * DR;
    float a = bdt.v[s][d];
#pragma unroll
    for (int r = 0; r < DR; ++r) a += xr[r] * wr[r];
    dt[idx] = (a > 20.0f) ? a : log1pf(expf(a));
}

// Selective scan: one thread per (stack,b,channel); 16 states in VGPRs,
// 512 sequential steps, no barriers in the loop. Output includes +D*xc and *silu(z).
__global__ void k_scan_ms(const float* __restrict__ dt, const float* __restrict__ xdbl,
                          const float* __restrict__ xc, const float* __restrict__ xz,
                          P4 Alog, P4 Dp, float* __restrict__ y)
{
    const int sb = blockIdx.x;               // 0 .. NSTK*Bc-1
    const int s  = sb / Bc;
    const int b  = sb % Bc;
    const int d  = threadIdx.x;              // 0..127
    float Ar[Ns], h[Ns];
    const float* al = Alog.v[s] + d * Ns;
#pragma unroll
    for (int n = 0; n < Ns; ++n) { Ar[n] = -expf(al[n]); h[n] = 0.0f; }
    const float Dd = Dp.v[s][d];
    const size_t rowBase = (size_t)s * BLc + (size_t)b * Lc;
    for (int t = 0; t < Lc; ++t) {
        const size_t row = rowBase + t;
        const float dtv  = dt[row * Din + d];
        const float xcv  = xc[row * Din + d];
        const float* Bs  = xdbl + row * XD + DR;
        const float* Cs  = Bs + Ns;
        const float dBxc = dtv * xcv;
        float acc = 0.0f;
#pragma unroll
        for (int n = 0; n < Ns; ++n) {
            const float dA = expf(dtv * Ar[n]);
            h[n] = dA * h[n] + dBxc * Bs[n];
            acc += h[n] * Cs[n];
        }
        const float z = xz[row * (2 * Din) + Din + d];
        y[row * Din + d] = (acc + Dd * xcv) * (z * sigmoidf_(z));
    }
}

// concat([fwd, flip(bwd)], -1): out is BLc x 2*Dm; bwd buffer is still reversed
__global__ void k_cat_stream(const float* __restrict__ fwd, const float* __restrict__ bwd,
                             float* __restrict__ out)
{
    const int idx = blockIdx.x * blockDim.x + threadIdx.x;
    if (idx >= BLc * 2 * Dm) return;
    const int c  = idx % (2 * Dm);
    const int bl = idx / (2 * Dm);
    const int b  = bl / Lc;
    const int l  = bl % Lc;
    out[idx] = (c < Dm)
        ? fwd[(size_t)bl * Dm + c]
        : bwd[((size_t)b * Lc + (Lc - 1 - l)) * Dm + (c - Dm)];
}

__global__ void k_concat2(const float* __restrict__ a, const float* __restrict__ b,
                          float* __restrict__ out)
{
    const int idx = blockIdx.x * blockDim.x + threadIdx.x;
    if (idx >= BLc * 2 * Dm) return;
    const int c   = idx % (2 * Dm);
    const int row = idx / (2 * Dm);
    out[idx] = (c < Dm) ? a[(size_t)row * Dm + c] : b[(size_t)row * Dm + (c - Dm)];
}

// h = sigmoid(g)*h + (1-sigmoid(g))*f   (in place on h)
__global__ void k_combine(float* __restrict__ h, const float* __restrict__ g,
                          const float* __restrict__ f)
{
    const int idx = blockIdx.x * blockDim.x + threadIdx.x;
    if (idx >= BLc * 2 * Dm) return;
    const float gg = sigmoidf_(g[idx]);
    h[idx] = gg * h[idx] + (1.0f - gg) * f[idx];
}

// masked mean-pool over L (mask = x[:,:,3] > 0.5)
__global__ void k_pool(const float* __restrict__ X, const float* __restrict__ h,
                       float* __restrict__ pooled)
{
    const int b = blockIdx.x;
    const int d = threadIdx.x;               // 0..127
    float sum = 0.0f, cnt = 0.0f;
    for (int t = 0; t < Lc; ++t) {
        const float m = (X[((size_t)b * Lc + t) * 4 + 3] > 0.5f) ? 1.0f : 0.0f;
        sum += m * h[((size_t)b * Lc + t) * (2 * Dm) + d];
        cnt += m;
    }
    pooled[b * (2 * Dm) + d] = sum / fmaxf(cnt, 1e-6f);
}

// final projection: out = pooled @ proj_W^T + proj_b   (8x128 @ 128x128, tiny)
__global__ void k_final(const float* __restrict__ pooled, const float* __restrict__ W,
                        const float* __restrict__ bb, float* __restrict__ out)
{
    const int idx = blockIdx.x * blockDim.x + threadIdx.x;
    if (idx >= Bc * 128) return;
    const int o = idx % 128;
    const int b = idx / 128;
    const float* pr = pooled + b * 128;
    const float* wr = W + (size_t)o * 128;
    float a = bb[o];
    for (int k = 0; k < 128; ++k) a += pr[k] * wr[k];
    out[idx] = a;
}

// ---------------------------------------------------------------------------
// Host launch
// ---------------------------------------------------------------------------
extern "C" void kernel_launch(void* const* d_in, const int* in_sizes, int n_in,
                              void* d_out, int out_size, void* d_ws, size_t ws_size,
                              hipStream_t stream)
{
    (void)in_sizes; (void)n_in; (void)out_size; (void)ws_size;
    auto in = [&](int k) { return (const float*)d_in[k]; };

    const float* X        = in(0);
    const float* len_W    = in(1);  const float* len_b    = in(2);
    const float* len_g    = in(3);  const float* len_be   = in(4);
    const float* cont_W   = in(5);  const float* cont_b   = in(6);
    const float* cont_g   = in(7);  const float* cont_be  = in(8);
    const float* dir_emb  = in(9);
    const int stackBase[NSTK] = {10, 23, 36, 49};   // len_fwd, len_bwd, struct_fwd, struct_bwd
    const float* len_proj_W    = in(62); const float* len_proj_b    = in(63);
    const float* struct_proj_W = in(64); const float* struct_proj_b = in(65);
    const float* gate_W   = in(66); const float* gate_b   = in(67);
    const float* fuse_W   = in(68); const float* fuse_b   = in(69);
    const float* proj_W   = in(70); const float* proj_b   = in(71);

    // leaf order in each stack: ln_g, ln_b, W_in, b_in, conv_w, conv_b,
    //                           W_x, W_dt, b_dt, A_log, D, W_out, b_out
    static const int leafStride[13] = {
        Dm, Dm, 2 * Din * Dm, 2 * Din, Din * 4, Din,
        XD * Din, Din * DR, Din, Din * Ns, Din, Dm * Din, Dm
    };
    auto leaf = [&](int stk, int li, int layer) {
        return (const float*)d_in[stackBase[stk] + li] + (size_t)layer * leafStride[li];
    };
    auto p4 = [&](int li, int layer) {
        P4 p; for (int s = 0; s < NSTK; ++s) p.v[s] = leaf(s, li, layer); return p;
    };
    auto p1 = [&](const float* p) {
        P4 q; q.v[0] = q.v[1] = q.v[2] = q.v[3] = p; return q;
    };

    // ---- workspace carve-up (floats) ----
    float* w = (float*)d_ws;
    size_t off = 0;
    auto alloc = [&](size_t nel) { float* p = w + off; off += nel; return p; };
    float* xlen  = alloc((size_t)BLc * Dm);
    float* xstr  = alloc((size_t)BLc * Dm);
    float* bufA  = alloc((size_t)NSTK * BLc * Dm);       // stream state (residual)
    float* bufLN = alloc((size_t)NSTK * BLc * Dm);
    float* xzb   = alloc((size_t)NSTK * BLc * 2 * Din);
    float* xcb   = alloc((size_t)NSTK * BLc * Din);
    float* xdb   = alloc((size_t)NSTK * BLc * XD);
    float* dtb   = alloc((size_t)NSTK * BLc * Din);
    float* ybuf  = alloc((size_t)NSTK * BLc * Din);
    float* catL  = alloc((size_t)BLc * 2 * Dm);
    float* catS  = alloc((size_t)BLc * 2 * Dm);
    float* hlen  = alloc((size_t)BLc * Dm);
    float* hstr  = alloc((size_t)BLc * Dm);
    float* hbig  = alloc((size_t)BLc * 2 * Dm);
    float* gbuf  = alloc((size_t)BLc * 2 * Dm);
    float* fbuf  = alloc((size_t)BLc * 2 * Dm);
    float* pooled = alloc((size_t)Bc * 2 * Dm);

    // ---- embedding + stack packing ----
    k_embed<<<BLc, 64, 0, stream>>>(X, len_W, len_b, len_g, len_be,
                                    cont_W, cont_b, cont_g, cont_be, dir_emb,
                                    xlen, xstr);
    {
        const int tot = NSTK * BLc * Dm;
        k_pack<<<(tot + 255) / 256, 256, 0, stream>>>(xlen, xstr, bufA);
    }

    // ---- 2 mamba layers, 4 stacks fused per launch ----
    for (int l = 0; l < 2; ++l) {
        k_ln_ms<<<NSTK * BLc, 64, 0, stream>>>(bufA, p4(0, l), p4(1, l), bufLN, BLc);
        {   // xz = ln(x) @ W_in^T + b_in   (M=16384, N=256, K=64)
            dim3 g((2 * Din) / 16, (NSTK * BLc) / 16);
            k_gemm_wmma_ms<<<g, 32, 0, stream>>>(bufLN, p4(2, l), p4(3, l),
                                                 nullptr, xzb, 2 * Din, Dm, BLc, 1);
        }
        {   const int tot = NSTK * BLc * Din;
            k_conv_silu_ms<<<(tot + 255) / 256, 256, 0, stream>>>(xzb, p4(4, l), p4(5, l), xcb);
        }
        {   const int tot = NSTK * BLc * XD;
            k_xdbl_ms<<<(tot + 255) / 256, 256, 0, stream>>>(xcb, p4(6, l), xdb);
        }
        {   const int tot = NSTK * BLc * Din;
            k_dt_ms<<<(tot + 255) / 256, 256, 0, stream>>>(xdb, p4(7, l), p4(8, l), dtb);
        }
        k_scan_ms<<<NSTK * Bc, Din, 0, stream>>>(dtb, xdb, xcb, xzb, p4(9, l), p4(10, l), ybuf);
        {   // x = res + y @ W_out^T + b_out   (M=16384, N=64, K=128), in-place residual
            dim3 g(Dm / 16, (NSTK * BLc) / 16);
            k_gemm_wmma_ms<<<g, 32, 0, stream>>>(ybuf, p4(11, l), p4(12, l),
                                                 bufA, bufA, Dm, Din, BLc, 1);
        }
    }

    // ---- unpack + fwd/bwd concat per stream ----
    {
        const int tot = BLc * 2 * Dm;
        k_cat_stream<<<(tot + 255) / 256, 256, 0, stream>>>(
            bufA + (size_t)0 * BLc * Dm, bufA + (size_t)1 * BLc * Dm, catL);
        k_cat_stream<<<(tot + 255) / 256, 256, 0, stream>>>(
            bufA + (size_t)2 * BLc * Dm, bufA + (size_t)3 * BLc * Dm, catS);
    }

    // ---- per-stream projections (K=128, N=64) ----
    {
        dim3 g(Dm / 16, BLc / 16);
        k_gemm_wmma_ms<<<g, 32, 0, stream>>>(catL, p1(len_proj_W), p1(len_proj_b),
                                             nullptr, hlen, Dm, 2 * Dm, BLc, 1);
        k_gemm_wmma_ms<<<g, 32, 0, stream>>>(catS, p1(struct_proj_W), p1(struct_proj_b),
                                             nullptr, hstr, Dm, 2 * Dm, BLc, 1);
    }
    {
        const int tot = BLc * 2 * Dm;
        k_concat2<<<(tot + 255) / 256, 256, 0, stream>>>(hlen, hstr, hbig);
    }

    // ---- gate / fuse (K=128, N=128) + combine ----
    {
        dim3 g((2 * Dm) / 16, BLc / 16);
        k_gemm_wmma_ms<<<g, 32, 0, stream>>>(hbig, p1(gate_W), p1(gate_b),
                                             nullptr, gbuf, 2 * Dm, 2 * Dm, BLc, 1);
        k_gemm_wmma_ms<<<g, 32, 0, stream>>>(hbig, p1(fuse_W), p1(fuse_b),
                                             nullptr, fbuf, 2 * Dm, 2 * Dm, BLc, 1);
    }
    {
        const int tot = BLc * 2 * Dm;
        k_combine<<<(tot + 255) / 256, 256, 0, stream>>>(hbig, gbuf, fbuf);
    }

    // ---- masked mean pool + final projection ----
    k_pool<<<Bc, 128, 0, stream>>>(X, hbig, pooled);
    {
        const int tot = Bc * 128;
        k_final<<<(tot + 255) / 256, 256, 0, stream>>>(pooled, proj_W, proj_b, (float*)d_out);
    }
}